// GyroplaneLayer_73632919322887
// MI455X (gfx1250) — compile-verified
//
#include <hip/hip_runtime.h>
#include <math.h>

typedef __attribute__((ext_vector_type(2))) float v2f;
typedef __attribute__((ext_vector_type(8))) float v8f;

#define B_DIM 4096
#define U_DIM 512
#define D_DIM 128
#define EPS1f 1e-5f
#define EPS2f 1e-8f

// D = A(16x4 f32) * B(4x16 f32) + C(16x16 f32), wave32 CDNA5 WMMA.
__device__ __forceinline__ v8f wmma4(v2f a, v2f b, v8f c) {
  return __builtin_amdgcn_wmma_f32_16x16x4_f32(
      /*neg_a=*/false, a, /*neg_b=*/false, b,
      /*c_mod=*/(short)0, c, /*reuse_a=*/false, /*reuse_b=*/false);
}

// ---------------- row statistics + transposed copies of p ----------------
__global__ __launch_bounds__(128) void gyro_stats_p(
    const float* __restrict__ p, const float* __restrict__ a,
    float* __restrict__ udu, float* __restrict__ na,
    float* __restrict__ p2, float* __restrict__ pa,
    float* __restrict__ pT, float* __restrict__ p2T) {
  __shared__ float s1[128], s2[128];
  const int u = blockIdx.x, d = threadIdx.x;
  const float pv = p[(size_t)u * D_DIM + d];
  const float av = a[(size_t)u * D_DIM + d];
  const float pv2 = pv * pv;
  p2[(size_t)u * D_DIM + d] = pv2;
  pa[(size_t)u * D_DIM + d] = pv * av;
  pT[(size_t)d * U_DIM + u] = pv;    // D x U layout for phase2 B-operand
  p2T[(size_t)d * U_DIM + u] = pv2;  // (one-time 512KB; makes phase2 loads b64)
  s1[d] = pv2;
  s2[d] = av * av;
  __syncthreads();
  for (int off = 64; off > 0; off >>= 1) {
    if (d < off) { s1[d] += s1[d + off]; s2[d] += s2[d + off]; }
    __syncthreads();
  }
  if (d == 0) { udu[u] = s1[0]; na[u] = sqrtf(s2[0]); }
}

__global__ __launch_bounds__(128) void gyro_stats_x(
    const float* __restrict__ x, float* __restrict__ vdv) {
  __shared__ float s1[128];
  const int b = blockIdx.x, d = threadIdx.x;
  const float xv = x[(size_t)b * D_DIM + d];
  s1[d] = xv * xv;
  __syncthreads();
  for (int off = 64; off > 0; off >>= 1) {
    if (d < off) s1[d] += s1[d + off];
    __syncthreads();
  }
  if (d == 0) vdv[b] = s1[0];
}

// ---------------- phase 1: G = X * P^T, derive A[b,u], B2[b,u] ----------------
// grid (B/128, U/32), 128 threads = 4 waves, wave -> 32x32 block (2x2 tiles)
__global__ __launch_bounds__(128) void gyro_phase1(
    const float* __restrict__ x, const float* __restrict__ p,
    const float* __restrict__ vdv, const float* __restrict__ udu,
    float* __restrict__ Abuf, float* __restrict__ Bbuf) {
  const int lane = threadIdx.x & 31;
  const int wave = threadIdx.x >> 5;
  const int hi = lane >> 4;
  const int lo = lane & 15;
  const int kh = hi << 1;
  const int m0 = blockIdx.x * 128 + wave * 32;
  const int n0 = blockIdx.y * 32;

  v8f c00 = {}, c01 = {}, c10 = {}, c11 = {};
  const float* xr0 = x + (size_t)(m0 + lo) * D_DIM;
  const float* xr1 = x + (size_t)(m0 + 16 + lo) * D_DIM;
  const float* pr0 = p + (size_t)(n0 + lo) * D_DIM;
  const float* pr1 = p + (size_t)(n0 + 16 + lo) * D_DIM;

  for (int k = 0; k < D_DIM; k += 4) {
    const int kk = k + kh;
    v2f a0 = *(const v2f*)(xr0 + kk);
    v2f a1 = *(const v2f*)(xr1 + kk);
    v2f b0 = *(const v2f*)(pr0 + kk);
    v2f b1 = *(const v2f*)(pr1 + kk);
    c00 = wmma4(a0, b0, c00);
    c01 = wmma4(a0, b1, c01);
    c10 = wmma4(a1, b0, c10);
    c11 = wmma4(a1, b1, c11);
  }

#pragma unroll
  for (int mt = 0; mt < 2; ++mt) {
#pragma unroll
    for (int nt = 0; nt < 2; ++nt) {
      v8f acc = (mt == 0) ? (nt == 0 ? c00 : c01) : (nt == 0 ? c10 : c11);
      const int ug = n0 + nt * 16 + lo;
      const float uduv = udu[ug];
#pragma unroll
      for (int i = 0; i < 8; ++i) {
        const int bg = m0 + mt * 16 + hi * 8 + i;
        const float G = acc[i];           // x . p  -> udv = -G
        const float vv = vdv[bg];
        const float two_udv = -2.f * G;
        const float alpha = 1.f + two_udv + vv;       // 1 + 2c udv + c vdv
        const float beta  = 1.f - uduv;               // 1 - c udu
        const float den   = 1.f + two_udv + uduv * vv + EPS2f;
        const float inv   = 1.f / den;
        Abuf[(size_t)bg * U_DIM + ug] = alpha * inv;
        Bbuf[(size_t)bg * U_DIM + ug] = beta * inv;
      }
    }
  }
}

// ---------------- phase 2: clip-scale s[b,d] ----------------
// norm2[b,d] = (A^2 . p^2)[b,d] - 2 x[b,d] (A*B2 . p)[b,d] + x^2 * sum_u B2^2
// grid (B/64, D/16), 128 threads = 4 waves, wave -> 16x16 tile, K = U = 512
// B-operands read from transposed pT/p2T (D x U) so per-lane loads are b64.
__global__ __launch_bounds__(128) void gyro_phase2(
    const float* __restrict__ x,
    const float* __restrict__ pT, const float* __restrict__ p2T,
    const float* __restrict__ Abuf, const float* __restrict__ Bbuf,
    float* __restrict__ sbuf) {
  const int lane = threadIdx.x & 31;
  const int wave = threadIdx.x >> 5;
  const int hi = lane >> 4;
  const int lo = lane & 15;
  const int kh = hi << 1;
  const int m0 = blockIdx.x * 64 + wave * 16;
  const int n0 = blockIdx.y * 16;

  v8f cM1 = {}, cM2 = {};
  float sb = 0.f;
  const float* Ar = Abuf + (size_t)(m0 + lo) * U_DIM;
  const float* Br = Bbuf + (size_t)(m0 + lo) * U_DIM;
  const float* b1r = p2T + (size_t)(n0 + lo) * U_DIM;  // B[k,n] = p2T[n][k]
  const float* b2r = pT + (size_t)(n0 + lo) * U_DIM;   // B[k,n] = pT[n][k]

  for (int k = 0; k < U_DIM; k += 4) {
    const int kk = k + kh;
    v2f Av = *(const v2f*)(Ar + kk);
    v2f Bv = *(const v2f*)(Br + kk);
    v2f aA2 = Av * Av;
    v2f aAB = Av * Bv;
    sb += Bv.x * Bv.x + Bv.y * Bv.y;
    v2f b1 = *(const v2f*)(b1r + kk);
    v2f b2 = *(const v2f*)(b2r + kk);
    cM1 = wmma4(aA2, b1, cM1);
    cM2 = wmma4(aAB, b2, cM2);
  }
  sb += __shfl_xor(sb, 16, 32);  // lane L now holds sum_u B2[m0+(L&15),u]^2

  const float clipn = 1.f - EPS1f;  // (1-eps1)/sqrt(c), c=1
#pragma unroll
  for (int i = 0; i < 8; ++i) {
    const int bg = m0 + hi * 8 + i;
    const int dg = n0 + lo;
    const float sbr = __shfl(sb, hi * 8 + i, 32);
    const float xv = x[(size_t)bg * D_DIM + dg];
    float n2 = cM1[i] - 2.f * xv * cM2[i] + xv * xv * sbr;
    n2 = fmaxf(n2, 0.f);
    const float nrm = sqrtf(n2);
    sbuf[(size_t)bg * D_DIM + dg] = clipn / fmaxf(nrm, clipn);
  }
}

// ---------------- phase 3: four GEMMs (K=D) + asinh epilogue ----------------
// grid (B/64, U/16), wave -> 16x16 tile with 4 accumulators
__global__ __launch_bounds__(128) void gyro_phase3(
    const float* __restrict__ x, const float* __restrict__ p,
    const float* __restrict__ a, const float* __restrict__ p2,
    const float* __restrict__ pa,
    const float* __restrict__ Abuf, const float* __restrict__ Bbuf,
    const float* __restrict__ sbuf, const float* __restrict__ na,
    float* __restrict__ out) {
  const int lane = threadIdx.x & 31;
  const int wave = threadIdx.x >> 5;
  const int hi = lane >> 4;
  const int lo = lane & 15;
  const int kh = hi << 1;
  const int m0 = blockIdx.x * 64 + wave * 16;
  const int n0 = blockIdx.y * 16;

  v8f cR1 = {}, cR2 = {}, cQ1 = {}, cQ2 = {};
  float q3 = 0.f;
  const float* sr = sbuf + (size_t)(m0 + lo) * D_DIM;
  const float* xr = x + (size_t)(m0 + lo) * D_DIM;
  const float* par = pa + (size_t)(n0 + lo) * D_DIM;
  const float* ar = a + (size_t)(n0 + lo) * D_DIM;
  const float* p2r = p2 + (size_t)(n0 + lo) * D_DIM;
  const float* pr = p + (size_t)(n0 + lo) * D_DIM;

  for (int k = 0; k < D_DIM; k += 4) {
    const int kk = k + kh;
    v2f sv = *(const v2f*)(sr + kk);
    v2f xv = *(const v2f*)(xr + kk);
    v2f aS = sv;              // s
    v2f aSX = sv * xv;        // s*x
    v2f aS2 = sv * sv;        // s^2
    v2f aS2X = aS2 * xv;      // s^2*x
    q3 += aS2.x * xv.x * xv.x + aS2.y * xv.y * xv.y;  // sum_d s^2 x^2
    v2f bPA = *(const v2f*)(par + kk);
    v2f bA  = *(const v2f*)(ar + kk);
    v2f bP2 = *(const v2f*)(p2r + kk);
    v2f bP  = *(const v2f*)(pr + kk);
    cR1 = wmma4(aS,   bPA, cR1);   // sum_d s * (p.a)
    cR2 = wmma4(aSX,  bA,  cR2);   // sum_d (s x) * a
    cQ1 = wmma4(aS2,  bP2, cQ1);   // sum_d s^2 * p^2
    cQ2 = wmma4(aS2X, bP,  cQ2);   // sum_d (s^2 x) * p
  }
  q3 += __shfl_xor(q3, 16, 32);

  const int ug = n0 + lo;
  const float nav = na[ug];
#pragma unroll
  for (int i = 0; i < 8; ++i) {
    const int bg = m0 + hi * 8 + i;
    const float q3r = __shfl(q3, hi * 8 + i, 32);
    const float Av = Abuf[(size_t)bg * U_DIM + ug];
    const float Bv = Bbuf[(size_t)bg * U_DIM + ug];
    const float moba = -Av * cR1[i] + Bv * cR2[i];
    const float nms = Av * Av * cQ1[i] - 2.f * Av * Bv * cQ2[i] + Bv * Bv * q3r;
    const float num = 2.f * fabsf(moba);
    const float den = (1.f - nms) * nav + EPS1f;
    const float z = num / den;
    out[(size_t)bg * U_DIM + ug] = logf(z + sqrtf(z * z + 1.f));  // asinh
  }
}

extern "C" void kernel_launch(void* const* d_in, const int* in_sizes, int n_in,
                              void* d_out, int out_size, void* d_ws, size_t ws_size,
                              hipStream_t stream) {
  (void)in_sizes; (void)n_in; (void)out_size; (void)ws_size;
  const float* x = (const float*)d_in[0];
  const float* p = (const float*)d_in[1];
  const float* a = (const float*)d_in[2];
  float* out = (float*)d_out;

  float* ws = (float*)d_ws;
  float* Abuf = ws;                                   // B*U
  float* Bbuf = Abuf + (size_t)B_DIM * U_DIM;         // B*U
  float* sbuf = Bbuf + (size_t)B_DIM * U_DIM;         // B*D
  float* vdv  = sbuf + (size_t)B_DIM * D_DIM;         // B
  float* udu  = vdv + B_DIM;                          // U
  float* na   = udu + U_DIM;                          // U
  float* p2   = na + U_DIM;                           // U*D
  float* pa   = p2 + (size_t)U_DIM * D_DIM;           // U*D
  float* pT   = pa + (size_t)U_DIM * D_DIM;           // D*U (transposed)
  float* p2T  = pT + (size_t)U_DIM * D_DIM;           // D*U (transposed)

  gyro_stats_p<<<U_DIM, 128, 0, stream>>>(p, a, udu, na, p2, pa, pT, p2T);
  gyro_stats_x<<<B_DIM, 128, 0, stream>>>(x, vdv);
  gyro_phase1<<<dim3(B_DIM / 128, U_DIM / 32), 128, 0, stream>>>(x, p, vdv, udu, Abuf, Bbuf);
  gyro_phase2<<<dim3(B_DIM / 64, D_DIM / 16), 128, 0, stream>>>(x, pT, p2T, Abuf, Bbuf, sbuf);
  gyro_phase3<<<dim3(B_DIM / 64, U_DIM / 16), 128, 0, stream>>>(x, p, a, p2, pa, Abuf, Bbuf, sbuf, na, out);
}